// HashEncoder_52759378264699
// MI455X (gfx1250) — compile-verified
//
#include <hip/hip_runtime.h>
#include <stdint.h>

// ------------------------------------------------------------------
// Multiresolution hash-grid encoder (Instant-NGP style), MI455X/gfx1250.
//
// Key observation: the reference uses HASH_MOD = TABLE_SIZES[0] = 4096 for
// ALL levels, so only the first 4096 entries (32KB) of each table are hot.
// We DMA levels 0..7 (256KB) into LDS with the CDNA5 Tensor Data Mover,
// overlapping the DMA with the global-table levels 8..15, then gather the
// coarse levels from LDS.
// ------------------------------------------------------------------

#define NUM_LEVELS   16
#define LDS_LEVELS   8            // levels 0..7 cached in LDS (8 * 32KB = 256KB)
#define HASH_MASK    4095u        // HASH_MOD = 16^3 = 4096 (reference quirk)
#define PRIME_Y      2654435761u
#define PRIME_Z      805459861u
#define THREADS      512          // 16 wave32s per workgroup
#define PPT          4            // points per thread

#if defined(__has_builtin)
#  if __has_builtin(__builtin_amdgcn_tensor_load_to_lds)
#    define HAVE_TDM 1
#  endif
#  if __has_builtin(__builtin_amdgcn_s_wait_tensorcnt)
#    define HAVE_WAIT_TENSOR 1
#  endif
#endif
#ifndef HAVE_TDM
#  define HAVE_TDM 0
#endif
#ifndef HAVE_WAIT_TENSOR
#  define HAVE_WAIT_TENSOR 0
#endif

struct Tabs { const float* t[NUM_LEVELS]; };

typedef __attribute__((ext_vector_type(4))) unsigned int u32x4_t;
typedef __attribute__((ext_vector_type(4))) int          i32x4_t;
typedef __attribute__((ext_vector_type(8))) int          i32x8_t;

__device__ __forceinline__ int iclamp(int v, int hi) {
  v = v < 0 ? 0 : v;
  return v > hi ? hi : v;
}

__device__ __forceinline__ float2 lerp2(float2 a, float2 b, float t) {
  float u = 1.0f - t;
  float2 r;
  r.x = a.x * u + b.x * t;
  r.y = a.y * u + b.y * t;
  return r;
}

// idx[c] for c = dx*4 + dy*2 + dz (matches the reference OFFSETS ordering).
__device__ __forceinline__ void level_hash(float px, float py, float pz, int lvl,
                                           unsigned idx[8],
                                           float& wx, float& wy, float& wz) {
  const int res = 16 << lvl;                 // RESOLUTIONS[lvl] (exact powers of 2)
  const float rm1 = (float)(res - 1);
  float sx = (px + 1.0f) * 0.5f * rm1;
  float sy = (py + 1.0f) * 0.5f * rm1;
  float sz = (pz + 1.0f) * 0.5f * rm1;
  float fx = floorf(sx), fy = floorf(sy), fz = floorf(sz);
  int gx = (int)fx, gy = (int)fy, gz = (int)fz;
  wx = sx - fx; wy = sy - fy; wz = sz - fz;
  const int hi = res - 1;
  unsigned hx0 = (unsigned)iclamp(gx,     hi);
  unsigned hx1 = (unsigned)iclamp(gx + 1, hi);
  unsigned hy0 = (unsigned)iclamp(gy,     hi) * PRIME_Y;
  unsigned hy1 = (unsigned)iclamp(gy + 1, hi) * PRIME_Y;
  unsigned hz0 = (unsigned)iclamp(gz,     hi) * PRIME_Z;
  unsigned hz1 = (unsigned)iclamp(gz + 1, hi) * PRIME_Z;
  idx[0] = (hx0 ^ hy0 ^ hz0) & HASH_MASK;
  idx[1] = (hx0 ^ hy0 ^ hz1) & HASH_MASK;
  idx[2] = (hx0 ^ hy1 ^ hz0) & HASH_MASK;
  idx[3] = (hx0 ^ hy1 ^ hz1) & HASH_MASK;
  idx[4] = (hx1 ^ hy0 ^ hz0) & HASH_MASK;
  idx[5] = (hx1 ^ hy0 ^ hz1) & HASH_MASK;
  idx[6] = (hx1 ^ hy1 ^ hz0) & HASH_MASK;
  idx[7] = (hx1 ^ hy1 ^ hz1) & HASH_MASK;
}

// Reference quirk: wx lerps the dz axis, wy the dy axis, wz the dx axis.
__device__ __forceinline__ float2 trilerp(const float2 f[8],
                                          float wx, float wy, float wz) {
  float2 c00 = lerp2(f[0], f[1], wx);
  float2 c01 = lerp2(f[2], f[3], wx);
  float2 c10 = lerp2(f[4], f[5], wx);
  float2 c11 = lerp2(f[6], f[7], wx);
  float2 c0  = lerp2(c00, c01, wy);
  float2 c1  = lerp2(c10, c11, wy);
  return lerp2(c0, c1, wz);
}

#if HAVE_TDM
// One TDM descriptor: 1-D copy of 8192 x 4B (= one level's hot 4096 x float2)
// from global memory into LDS. D# layout per CDNA5 ISA ch.8.
__device__ __forceinline__ void tdm_load_32k(const void* gsrc, unsigned lds_byte) {
  unsigned long long ga = (unsigned long long)(uintptr_t)gsrc;
  const unsigned NE = 8192u;                       // 4-byte elements per level
  u32x4_t g0;
  g0.x = 1u;                                       // count=1 valid, is_restore=0, gather off
  g0.y = lds_byte;                                 // D#.lds_addr (bytes)
  g0.z = (unsigned)ga;                             // D#.global_addr[31:0]
  g0.w = (unsigned)((ga >> 32) & 0x01FFFFFFu)      // D#.global_addr[56:32]
       | (2u << 30);                               // type = 2 ("image")
  i32x8_t g1;
  g1[0] = (int)(2u << 16);                         // workgroup_mask=0, data_size=2 (4B)
  g1[1] = (int)((NE & 0xFFFFu) << 16);             // tensor_dim0[15:0]
  g1[2] = (int)((NE >> 16) | (1u << 16));          // tensor_dim0[31:16], tensor_dim1=1
  g1[3] = (int)(NE << 16);                         // tile_dim0 = 8192
  g1[4] = 0;                                       // tile_dim1 = tile_dim2 = 0 (unused)
  g1[5] = (int)NE;                                 // tensor_dim0_stride (lo 32)
  g1[6] = 0;
  g1[7] = 0;
  i32x4_t gz4 = (i32x4_t)0;
#if defined(__clang_major__) && __clang_major__ >= 23
  i32x8_t gz8 = (i32x8_t)0;
  __builtin_amdgcn_tensor_load_to_lds(g0, g1, gz4, gz4, gz8, 0);
#else
  __builtin_amdgcn_tensor_load_to_lds(g0, g1, gz4, gz4, 0);
#endif
}
#endif

__global__ __launch_bounds__(THREADS)
void HashEncoder_gfx1250_kernel(const float* __restrict__ pos, Tabs tabs,
                                float* __restrict__ out, int n) {
  __shared__ float2 lds_tab[LDS_LEVELS * 4096];    // 256 KB

  // ---- Stage 1: kick off the TDM DMA of levels 0..7 into LDS ----
#if HAVE_TDM
  if (threadIdx.x == 0) {
    unsigned base = (unsigned)(uintptr_t)(&lds_tab[0]);
    #pragma unroll
    for (int l = 0; l < LDS_LEVELS; ++l)
      tdm_load_32k(tabs.t[l], base + (unsigned)(l * 32768));
  }
#else
  for (int e = (int)threadIdx.x; e < LDS_LEVELS * 4096; e += THREADS) {
    int l = e >> 12;
    lds_tab[e] = ((const float2*)tabs.t[l])[e & 4095];
  }
#endif

  const int tid0   = (int)(blockIdx.x * THREADS + threadIdx.x);
  const int stride = (int)(gridDim.x * THREADS);
  float2* __restrict__ out2 = (float2*)out;        // 16 float2 per point row

  float px[PPT], py[PPT], pz[PPT];
  #pragma unroll
  for (int k = 0; k < PPT; ++k) {
    int i = tid0 + k * stride;
    if (i < n) { px[k] = pos[3*i]; py[k] = pos[3*i+1]; pz[k] = pos[3*i+2]; }
    else       { px[k] = 0.0f; py[k] = 0.0f; pz[k] = 0.0f; }
  }

  // ---- Phase A: fine levels (8..15) from L2-resident tables.
  //      This overlaps with the in-flight TDM DMA. ----
  #pragma unroll
  for (int k = 0; k < PPT; ++k) {
    int i = tid0 + k * stride;
    if (i >= n) continue;
    for (int lvl = LDS_LEVELS; lvl < NUM_LEVELS; ++lvl) {
      unsigned idx[8]; float wx, wy, wz;
      level_hash(px[k], py[k], pz[k], lvl, idx, wx, wy, wz);
      const float2* __restrict__ t = (const float2*)tabs.t[lvl];
      float2 f[8];
      #pragma unroll
      for (int c = 0; c < 8; ++c) f[c] = t[idx[c]];
      out2[(size_t)i * NUM_LEVELS + lvl] = trilerp(f, wx, wy, wz);
    }
  }

  // ---- Wait for the TDM (per-wave TENSORcnt) then sync the workgroup ----
#if HAVE_TDM
  if (threadIdx.x == 0) {
#if HAVE_WAIT_TENSOR
    __builtin_amdgcn_s_wait_tensorcnt(0);
#else
    asm volatile("s_wait_tensorcnt 0x0" ::: "memory");
#endif
  }
#endif
  __syncthreads();

  // ---- Phase B: coarse levels (0..7) gathered from LDS ----
  #pragma unroll
  for (int k = 0; k < PPT; ++k) {
    int i = tid0 + k * stride;
    if (i >= n) continue;
    for (int lvl = 0; lvl < LDS_LEVELS; ++lvl) {
      unsigned idx[8]; float wx, wy, wz;
      level_hash(px[k], py[k], pz[k], lvl, idx, wx, wy, wz);
      const int b = lvl << 12;
      float2 f[8];
      #pragma unroll
      for (int c = 0; c < 8; ++c) f[c] = lds_tab[b + (int)idx[c]];
      out2[(size_t)i * NUM_LEVELS + lvl] = trilerp(f, wx, wy, wz);
    }
  }
}

extern "C" void kernel_launch(void* const* d_in, const int* in_sizes, int n_in,
                              void* d_out, int out_size, void* d_ws, size_t ws_size,
                              hipStream_t stream) {
  (void)out_size; (void)d_ws; (void)ws_size;

  const float* pos = (const float*)d_in[0];
  const int n = in_sizes[0] / 3;

  Tabs tabs;
  if (n_in >= 1 + NUM_LEVELS) {
    // tables passed as 16 separate inputs
    for (int l = 0; l < NUM_LEVELS; ++l) tabs.t[l] = (const float*)d_in[1 + l];
  } else {
    // fallback: tables concatenated flat in one buffer
    const float* base = (const float*)d_in[1];
    unsigned long long off = 0;
    for (int l = 0; l < NUM_LEVELS; ++l) {
      tabs.t[l] = base + off;
      unsigned long long r = 16ull << l;
      unsigned long long ts = r * r * r;
      if (ts > 524288ull) ts = 524288ull;            // min(res^3, 2^19)
      off += ts * 2ull;                              // LEVEL_DIM = 2
    }
  }

  const int per_block = THREADS * PPT;
  const int blocks = (n + per_block - 1) / per_block;
  hipLaunchKernelGGL(HashEncoder_gfx1250_kernel, dim3(blocks), dim3(THREADS), 0,
                     stream, pos, tabs, (float*)d_out, n);
}